// DetectHead_64776696758332
// MI455X (gfx1250) — compile-verified
//
#include <hip/hip_runtime.h>

// ---------------------------------------------------------------------------
// YOLO detect-head decode, fused over 3 scales. Pure streaming / HBM-bound:
// 204 MB total traffic -> ~8.8 us floor at 23.3 TB/s. No matmul exists in this
// op, so no WMMA; the win is perfect B128 coalescing + hardware exp/rcp.
// ---------------------------------------------------------------------------

constexpr int NB   = 64;      // batch
constexpr int NA   = 3;       // anchors per scale
constexpr int CH   = 14;      // 5 + 8 classes + 1 depth
constexpr int NTOT = 29484;   // 3*(48*156 + 24*78 + 12*39)

__device__ __forceinline__ float fast_rcp(float x) {
#if __has_builtin(__builtin_amdgcn_rcpf)
  return __builtin_amdgcn_rcpf(x);          // v_rcp_f32
#else
  return __fdividef(1.0f, x);
#endif
}

__device__ __forceinline__ float sigmoid_f(float x) {
  return fast_rcp(1.0f + __expf(-x));       // v_exp_f32 + v_rcp_f32
}

// W,H: feature-map dims. NOFS: offset of this scale in the concatenated N axis.
// AWi/AHi: integer anchor sizes for the 3 anchors of this scale.
// Note: stride*W == 1248 and stride*H == 384 for every scale.
template<int W, int H, int NOFS, int AW0, int AH0, int AW1, int AH1, int AW2, int AH2>
__device__ __forceinline__ void decode_scale(const float* __restrict__ in,
                                             float* __restrict__ out,
                                             unsigned u) {
  constexpr int HW = W * H;
  constexpr int Q  = HW / 4;                 // float4 chunks per (b, a) map

  const unsigned ba  = u / (unsigned)Q;      // b*3 + a   (const-divide -> mul/shift)
  const unsigned hw4 = u - ba * (unsigned)Q;
  const unsigned b   = ba / 3u;
  const unsigned a   = ba - b * 3u;

  // bw = exp(o2) * anchor_w/(stride*W);  bh = exp(o3) * anchor_h/(stride*H)
  const float aw = (a == 0u ? (float)AW0 : (a == 1u ? (float)AW1 : (float)AW2)) * (1.0f / 1248.0f);
  const float ah = (a == 0u ? (float)AH0 : (a == 1u ? (float)AH1 : (float)AH2)) * (1.0f / 384.0f);

  const unsigned hw0 = hw4 * 4u;
  const float* p = in + (size_t)ba * (size_t)(CH * HW) + hw0;

  __builtin_prefetch(p, 0, 0);               // global_prefetch_b8 (gfx1250)

  // 14 channel loads, each global_load_b128: a wave covers 512 contiguous bytes.
  float v[CH][4];
#pragma unroll
  for (int ch = 0; ch < CH; ++ch) {
    float4 ld = *reinterpret_cast<const float4*>(p + (size_t)ch * HW);
    v[ch][0] = ld.x; v[ch][1] = ld.y; v[ch][2] = ld.z; v[ch][3] = ld.w;
  }

  float* boxes = out;                                  // [NB, NTOT, 1, 4]
  float* confs = out + (size_t)NB * NTOT * 4;          // [NB, NTOT, 8]
  float* depth = out + (size_t)NB * NTOT * 12;         // [NB, NTOT, 1]

  const size_t gn0 = (size_t)b * NTOT + NOFS + (size_t)a * HW + hw0;  // multiple of 4

  float dep[4];
#pragma unroll
  for (int i = 0; i < 4; ++i) {
    const unsigned hwv = hw0 + (unsigned)i;
    const unsigned y   = hwv / (unsigned)W;            // const-divide -> mul/shift
    const unsigned x   = hwv - y * (unsigned)W;

    const float bx  = (sigmoid_f(v[0][i]) + (float)x) * (1.0f / (float)W);
    const float by  = (sigmoid_f(v[1][i]) + (float)y) * (1.0f / (float)H);
    const float bw  = __expf(v[2][i]) * aw;
    const float bh  = __expf(v[3][i]) * ah;
    const float det = sigmoid_f(v[4][i]);

    const float x1 = bx - 0.5f * bw;
    const float y1 = by - 0.5f * bh;

    const size_t gn = gn0 + (size_t)i;

    float4 bo; bo.x = x1; bo.y = y1; bo.z = x1 + bw; bo.w = y1 + bh;
    *reinterpret_cast<float4*>(boxes + gn * 4) = bo;   // 16B aligned

    float4 c0, c1;
    c0.x = sigmoid_f(v[5][i])  * det;  c0.y = sigmoid_f(v[6][i])  * det;
    c0.z = sigmoid_f(v[7][i])  * det;  c0.w = sigmoid_f(v[8][i])  * det;
    c1.x = sigmoid_f(v[9][i])  * det;  c1.y = sigmoid_f(v[10][i]) * det;
    c1.z = sigmoid_f(v[11][i]) * det;  c1.w = sigmoid_f(v[12][i]) * det;
    *reinterpret_cast<float4*>(confs + gn * 8)     = c0;  // 32B aligned
    *reinterpret_cast<float4*>(confs + gn * 8 + 4) = c1;

    dep[i] = v[13][i];
  }
  float4 dq; dq.x = dep[0]; dq.y = dep[1]; dq.z = dep[2]; dq.w = dep[3];
  *reinterpret_cast<float4*>(depth + gn0) = dq;        // 16B aligned, contiguous per wave
}

__global__ __launch_bounds__(256)
void detect_head_decode(const float* __restrict__ in1,
                        const float* __restrict__ in2,
                        const float* __restrict__ in3,
                        float* __restrict__ out) {
  // thread-range boundaries land exactly on 256-thread block boundaries for
  // scales 1 and 2 -> no intra-block scale divergence.
  constexpr unsigned E1 = (unsigned)NB * NA * (48 * 156 / 4);   // 359424
  constexpr unsigned E2 = E1 + (unsigned)NB * NA * (24 * 78 / 4); // 449280
  constexpr unsigned E3 = E2 + (unsigned)NB * NA * (12 * 39 / 4); // 471744

  const unsigned t = blockIdx.x * blockDim.x + threadIdx.x;
  if (t < E1) {
    decode_scale<156, 48, 0,      12,  16,  19,  36,  40,  28 >(in1, out, t);
  } else if (t < E2) {
    decode_scale< 78, 24, 22464,  36,  75,  76,  55,  72, 146 >(in2, out, t - E1);
  } else if (t < E3) {
    decode_scale< 39, 12, 28080, 142, 110, 192, 243, 459, 401 >(in3, out, t - E2);
  }
}

extern "C" void kernel_launch(void* const* d_in, const int* in_sizes, int n_in,
                              void* d_out, int out_size, void* d_ws, size_t ws_size,
                              hipStream_t stream) {
  (void)in_sizes; (void)n_in; (void)d_ws; (void)ws_size; (void)out_size;
  const float* in1 = (const float*)d_in[0];
  const float* in2 = (const float*)d_in[1];
  const float* in3 = (const float*)d_in[2];
  float* out = (float*)d_out;

  constexpr unsigned TOTAL = 471744u;   // sum of per-scale float4 chunks
  dim3 block(256);
  dim3 grid((TOTAL + 255u) / 256u);     // 1843 blocks
  hipLaunchKernelGGL(detect_head_decode, grid, block, 0, stream, in1, in2, in3, out);
}